// Sampler_11244224381384
// MI455X (gfx1250) — compile-verified
//
#include <hip/hip_runtime.h>
#include <stdint.h>

// ---------------------------------------------------------------------------
// Sampler for B=128 rows, V=128256 vocab (top-k <= 49, top-p, exp-noise draw,
// top-20 logprobs). One workgroup (512 threads = 16 wave32) per row.
//
// Pass 1 streams the row HBM -> LDS with double-buffered async copies
// (global_load_async_to_lds_b128 / s_wait_asynccnt, per-wave ASYNCcnt) while
// building a 4096-bin monotone-key histogram. Pass 2 (L2-resident: the whole
// 65.7MB logits tensor fits in the 192MB L2 after pass 1) gathers the few
// candidates above the k-th-largest bin. Everything after that is O(candidates).
// ---------------------------------------------------------------------------

constexpr int   kV      = 128256;
constexpr int   kV4     = kV / 4;       // 32064 float4 per row
constexpr int   kNT     = 512;          // threads per block
constexpr int   kNW     = kNT / 32;     // 16 waves
constexpr int   kChunks = kV4 / 32;     // 1002 chunks of 32 float4 per row (exact)
constexpr int   kBins   = 4096;
constexpr int   kMaxC   = 1024;         // candidate cap (typ. <= ~64)
constexpr float kEps    = 1e-5f;

__device__ __forceinline__ float fneginf() { return __int_as_float(0xff800000); }

// order-preserving 12-bit key from float bits
__device__ __forceinline__ unsigned fkey(float x) {
  unsigned b = __float_as_uint(x);
  unsigned m = (b & 0x80000000u) ? ~b : (b | 0x80000000u);
  return m >> 20;
}

__global__ __launch_bounds__(kNT) void sampler_kernel(
    const float* __restrict__ logits, const float* __restrict__ temperature,
    const int* __restrict__ top_k, const float* __restrict__ top_p,
    const float* __restrict__ u, int nl,
    int* __restrict__ out_sampled, int* __restrict__ out_idx,
    float* __restrict__ out_lp)
{
  const int row  = blockIdx.x;
  const int tid  = threadIdx.x;
  const int wave = tid >> 5;
  const int lane = tid & 31;

  __shared__ unsigned hist[kBins];     // 16 KB
  __shared__ float4   stage[2 * kNT];  // 16 KB async staging (per-wave slots)
  __shared__ float    cval[kMaxC];     // 4 KB
  __shared__ int      cidx[kMaxC];     // 4 KB
  __shared__ float    ev[kMaxC];       // 4 KB
  __shared__ float    rv[kNT];
  __shared__ int      ri[kNT];
  __shared__ int      sh_bin, sh_cnt, sh_m, sh_F;
  __shared__ float    sh_thresh, sh_logZF;

  for (int b = tid; b < kBins; b += kNT) hist[b] = 0u;
  for (int i = tid; i < kMaxC; i += kNT) { cval[i] = fneginf(); cidx[i] = 0x7fffffff; }
  if (tid == 0) { sh_cnt = 0; sh_m = 0; }
  __syncthreads();

  const float temp   = temperature[row];
  const bool  greedy = (temp < kEps);
  const float invt   = greedy ? 1.0f : (1.0f / temp);
  const float4* rowp = (const float4*)(logits + (size_t)row * kV);

  // ---- pass 1: async-stream row into LDS (double buffered, per-wave) ------
  {
    int cur = wave;   // this wave's chunk ids: wave, wave+16, ...
    int buf = 0;
    if (cur < kChunks) {
      uint64_t ga = (uint64_t)(uintptr_t)(rowp + (cur * 32 + lane));
      uint32_t la = (uint32_t)(uintptr_t)(&stage[buf * kNT + wave * 32 + lane]);
      asm volatile("global_load_async_to_lds_b128 %0, %1, off"
                   :: "v"(la), "v"(ga) : "memory");
    }
    while (cur < kChunks) {
      const int nxt = cur + kNW;
      if (nxt < kChunks) {
        uint64_t ga = (uint64_t)(uintptr_t)(rowp + (nxt * 32 + lane));
        uint32_t la = (uint32_t)(uintptr_t)(&stage[(buf ^ 1) * kNT + wave * 32 + lane]);
        asm volatile("global_load_async_to_lds_b128 %0, %1, off"
                     :: "v"(la), "v"(ga) : "memory");
        asm volatile("s_wait_asynccnt 1" ::: "memory"); // previous chunk landed
      } else {
        asm volatile("s_wait_asynccnt 0" ::: "memory"); // drain
      }
      float4 v = stage[buf * kNT + wave * 32 + lane];
      atomicAdd(&hist[fkey(v.x * invt)], 1u);
      atomicAdd(&hist[fkey(v.y * invt)], 1u);
      atomicAdd(&hist[fkey(v.z * invt)], 1u);
      atomicAdd(&hist[fkey(v.w * invt)], 1u);
      cur = nxt;
      buf ^= 1;
    }
  }
  __syncthreads();

  // ---- in-place parallel suffix sum: hist[b] := #elements with key >= b ----
  for (int d = 1; d < kBins; d <<= 1) {
    unsigned t[kBins / kNT];
#pragma unroll
    for (int e = 0; e < kBins / kNT; ++e) {
      const int b = tid + e * kNT;
      unsigned s = hist[b];
      if (b + d < kBins) s += hist[b + d];
      t[e] = s;
    }
    __syncthreads();
#pragma unroll
    for (int e = 0; e < kBins / kNT; ++e) hist[tid + e * kNT] = t[e];
    __syncthreads();
  }

  int K = top_k[row];
  if (K < 1) K = 1;                 // reference clips to [1, V]; here K <= 49
  // largest bin b with suffix-count >= K (unique: suffix sums are monotone)
#pragma unroll
  for (int e = 0; e < kBins / kNT; ++e) {
    const int b = tid + e * kNT;
    const unsigned c0 = hist[b];
    const unsigned c1 = (b + 1 < kBins) ? hist[b + 1] : 0u;
    if (c0 >= (unsigned)K && c1 < (unsigned)K) sh_bin = b;
  }
  __syncthreads();

  // ---- pass 2 (L2-resident re-scan): collect candidates above bin ---------
  const int binT = sh_bin;
  for (int i = tid; i < kV4; i += kNT) {
    const float4 v = rowp[i];
    const float xs[4] = { v.x * invt, v.y * invt, v.z * invt, v.w * invt };
#pragma unroll
    for (int e = 0; e < 4; ++e) {
      if ((int)fkey(xs[e]) >= binT) {
        const int pos = atomicAdd(&sh_cnt, 1);
        if (pos < kMaxC) { cval[pos] = xs[e]; cidx[pos] = i * 4 + e; }
      }
    }
  }
  __syncthreads();
  int C = sh_cnt; if (C > kMaxC) C = kMaxC;

  // ---- bitonic sort candidates: value desc, vocab-index asc on ties -------
  int P = 64; while (P < C) P <<= 1;          // power-of-two window, padded -inf
  for (int k = 2; k <= P; k <<= 1) {
    for (int j = k >> 1; j > 0; j >>= 1) {
      for (int i = tid; i < P; i += kNT) {
        const int ixj = i ^ j;
        if (ixj > i) {
          const float a = cval[i], b2 = cval[ixj];
          const int  ai = cidx[i], bi = cidx[ixj];
          const bool aFirst = (a > b2) || (a == b2 && ai < bi);
          const bool desc   = ((i & k) == 0);
          if (desc != aFirst) {
            cval[i] = b2; cval[ixj] = a;
            cidx[i] = bi; cidx[ixj] = ai;
          }
        }
      }
      __syncthreads();
    }
  }

  // ---- exact top-k threshold and kept count m -----------------------------
  const int kk = (K < C) ? K : C;
  if (tid == 0) sh_thresh = cval[kk - 1];
  __syncthreads();
  const float thr = sh_thresh;
  for (int i = tid; i < P; i += kNT)
    if (cval[i] >= thr) atomicMax(&sh_m, i + 1);
  __syncthreads();
  const int m = sh_m;

  // ---- softmax over kept set, top-p suffix cutoff, logZ -------------------
  const float v0 = cval[0];
  for (int i = tid; i < m; i += kNT) ev[i] = expf(cval[i] - v0);
  __syncthreads();
  if (tid == 0) {
    float Z = 0.f;
    for (int i = 0; i < m; ++i) Z += ev[i];
    const float target = top_p[row] * Z;     // drop d_j iff sum(e_0..e_{j-1}) >= p*Z
    float S = 0.f; int F = m;
    for (int j = 0; j < m; ++j) {
      if (j > 0 && S >= target) { F = j; break; }
      S += ev[j];
    }
    sh_F = F;
    sh_logZF = v0 + logf(S);                 // S == sum of kept e_i
  }
  __syncthreads();

  // ---- exponential-noise argmax over final kept set (gather u sparsely) ---
  const int F = sh_F;
  float bv = fneginf(); int bi = 0x7fffffff;
  for (int i = tid; i < F; i += kNT) {
    const int vi = cidx[i];
    const float uu = u[(size_t)row * kV + vi];
    const float q  = fmaxf(-log1pf(-uu), 1e-10f);
    const float s  = cval[i] - logf(q);
    if (s > bv || (s == bv && vi < bi)) { bv = s; bi = vi; }
  }
  rv[tid] = bv; ri[tid] = bi;
  __syncthreads();
  for (int s = kNT / 2; s > 0; s >>= 1) {
    if (tid < s) {
      if (rv[tid + s] > rv[tid] ||
          (rv[tid + s] == rv[tid] && ri[tid + s] < ri[tid])) {
        rv[tid] = rv[tid + s]; ri[tid] = ri[tid + s];
      }
    }
    __syncthreads();
  }

  // ---- outputs ------------------------------------------------------------
  if (tid == 0) {
    out_sampled[row] = greedy ? cidx[0] : ri[0];
    int*   oid = out_idx + (size_t)row * nl;
    float* olp = out_lp  + (size_t)row * nl;
    const float lZ = sh_logZF;
    int j = 0;
    for (; j < nl && j < F; ++j) { oid[j] = cidx[j]; olp[j] = cval[j] - lZ; }
    // pad with -inf logprobs at the smallest unused vocab indices (top_k ties)
    int v = 0;
    for (; j < nl; ++j) {
      for (;;) {
        bool used = false;
        for (int t = 0; t < F; ++t) if (cidx[t] == v) { used = true; break; }
        if (!used) break;
        ++v;
      }
      oid[j] = v++; olp[j] = fneginf();
    }
  }
}

extern "C" void kernel_launch(void* const* d_in, const int* in_sizes, int n_in,
                              void* d_out, int out_size, void* d_ws, size_t ws_size,
                              hipStream_t stream) {
  (void)n_in; (void)d_ws; (void)ws_size;
  const float* logits      = (const float*)d_in[0];
  const float* temperature = (const float*)d_in[1];
  const int*   top_k       = (const int*)d_in[2];
  const float* top_p       = (const float*)d_in[3];
  const float* u           = (const float*)d_in[4];
  const int B  = in_sizes[1];                    // 128 rows
  const int nl = (out_size - B) / (2 * B);       // 20 logprobs per row
  int*   out_sampled = (int*)d_out;
  int*   out_idx     = (int*)d_out + B;
  float* out_lp      = (float*)d_out + B + (size_t)B * nl;
  sampler_kernel<<<B, kNT, 0, stream>>>(logits, temperature, top_k, top_p, u, nl,
                                        out_sampled, out_idx, out_lp);
}